// GraphAttentionLayer_52020643889235
// MI455X (gfx1250) — compile-verified
//
#include <hip/hip_runtime.h>
#include <hip/hip_bf16.h>

typedef __attribute__((ext_vector_type(2))) float v2f;
typedef __attribute__((ext_vector_type(8))) float v8f;
typedef __attribute__((ext_vector_type(4))) int   v4i;

#define N_NODES 8192
#define F_IN    512
#define F_OUT   64
#define ALPHA   0.2f
#define TM      32      // rows of output per workgroup in attention kernel
#define TJ      64      // neighbor columns per LDS tile

// gfx1250 async global->LDS DMA (ASYNCcnt-tracked), with safe fallback.
// Builtin signature (from hipcc diagnostics): (v4i as1*, v4i as3*, Imm, Imm).
#if __has_builtin(__builtin_amdgcn_global_load_async_to_lds_b128)
#define USE_ASYNC_LDS 1
#if __has_builtin(__builtin_amdgcn_s_wait_asynccnt)
#define WAIT_ASYNC(n) __builtin_amdgcn_s_wait_asynccnt(n)
#else
#define WAIT_ASYNC(n) asm volatile("s_wait_asynccnt " #n ::: "memory")
#endif
#define ASYNC_CP16(gsrc, ldst)                                                 \
  __builtin_amdgcn_global_load_async_to_lds_b128(                              \
      (__attribute__((address_space(1))) v4i*)(gsrc),                          \
      (__attribute__((address_space(3))) v4i*)(ldst), 0, 0)
#else
#define USE_ASYNC_LDS 0
#define WAIT_ASYNC(n)
#endif

// ---------------------------------------------------------------------------
// Kernel A: h = X @ W   (f32 WMMA 16x16x4)
// grid = 512 blocks of 16 rows; 128 threads = 4 waves; wave w owns N-tile w.
// A-frag (32-bit A 16x4): lane<16 -> K={k0,k0+1}, lane>=16 -> K={k0+2,k0+3},
// M = lane&15.  B-frag mirrors A.  C/D: VGPR r holds rows r (lanes 0-15) and
// r+8 (lanes 16-31), N = lane&15.
// ---------------------------------------------------------------------------
__global__ __launch_bounds__(128)
void gat_h_kernel(const float* __restrict__ X, const float* __restrict__ W,
                  float* __restrict__ H) {
  const int mbase = blockIdx.x * 16;
  const int lane  = threadIdx.x & 31;
  const int wave  = threadIdx.x >> 5;
  const int nbase = wave * 16;
  const int lm    = lane & 15;
  const int kof   = (lane >> 4) * 2;

  const float* xrow = X + (size_t)(mbase + lm) * F_IN;
  v8f acc = {};
  for (int k0 = 0; k0 < F_IN; k0 += 4) {
    v2f a = *(const v2f*)(xrow + k0 + kof);           // 8B-aligned
    v2f b;
    b.x = W[(k0 + kof    ) * F_OUT + nbase + lm];
    b.y = W[(k0 + kof + 1) * F_OUT + nbase + lm];
    acc = __builtin_amdgcn_wmma_f32_16x16x4_f32(false, a, false, b,
                                                (short)0, acc, false, false);
  }
  const int rbase = mbase + ((lane < 16) ? 0 : 8);
  const int col   = nbase + lm;
#pragma unroll
  for (int r = 0; r < 8; ++r)
    H[(size_t)(rbase + r) * F_OUT + col] = acc[r];
}

// ---------------------------------------------------------------------------
// Kernel B: s1 = h@a1, s2 = h@a2, per-block max of s2 (deterministic reduce).
// One row per thread; 32 blocks x 256 threads.
// ---------------------------------------------------------------------------
__global__ __launch_bounds__(256)
void gat_s_kernel(const float* __restrict__ H, const float* __restrict__ a,
                  float* __restrict__ s1, float* __restrict__ s2,
                  float* __restrict__ blockmax) {
  __shared__ float red[256];
  const int row = blockIdx.x * 256 + threadIdx.x;
  const float* hr = H + (size_t)row * F_OUT;
  float d1 = 0.f, d2 = 0.f;
#pragma unroll 8
  for (int c = 0; c < F_OUT; ++c) {
    float hv = hr[c];
    d1 += hv * a[c];
    d2 += hv * a[F_OUT + c];
  }
  s1[row] = d1;
  s2[row] = d2;
  red[threadIdx.x] = d2;
  __syncthreads();
  for (int s = 128; s > 0; s >>= 1) {
    if (threadIdx.x < s)
      red[threadIdx.x] = fmaxf(red[threadIdx.x], red[threadIdx.x + s]);
    __syncthreads();
  }
  if (threadIdx.x == 0) blockmax[blockIdx.x] = red[0];
}

__global__ void gat_smax_kernel(const float* __restrict__ blockmax,
                                float* __restrict__ s2max) {
  if (threadIdx.x == 0) {
    float m = blockmax[0];
    for (int i = 1; i < 32; ++i) m = fmaxf(m, blockmax[i]);
    *s2max = m;   // shift bound: m_i = leaky(s1_i + s2max) >= all e_ij
  }
}

// ---------------------------------------------------------------------------
// Kernel C: fused masked-softmax(attention) @ h.  Single pass over adj (268MB)
// with double-buffered async global->LDS DMA.  grid = 256 blocks of TM=32
// rows; 256 threads = 8 waves; wave = (msub<<2)|ntile.  Never materializes
// the 8192x8192 attention matrix.
// ---------------------------------------------------------------------------
__global__ __launch_bounds__(256)
void gat_attn_kernel(const float* __restrict__ H, const int* __restrict__ adj,
                     const float* __restrict__ s1, const float* __restrict__ s2,
                     const float* __restrict__ s2max, float* __restrict__ out) {
  __shared__ int   adjT[2][TM][TJ];     // 16 KB
  __shared__ float hT[2][TJ][F_OUT];    // 32 KB
  __shared__ float pT[TM][TJ + 1];      // +1 pad: conflict-free A-frag reads
  __shared__ float mT[TM];
  __shared__ float zscr[TM][8];
  __shared__ float Zrow[TM];

  const int tid   = threadIdx.x;
  const int lane  = tid & 31;
  const int wave  = tid >> 5;
  const int msub  = wave >> 2;
  const int ntile = wave & 3;
  const int ibase = blockIdx.x * TM;
  const int lm    = lane & 15;
  const int kof   = (lane >> 4) * 2;

  // Stage one tile (TM x TJ adj ints, TJ x 64 h floats) into LDS buffer `buf`.
  // 6 x B128 per thread => 6 ASYNCcnt increments per wave per tile.
  auto issue_tile = [&](int buf, int jbase) {
#pragma unroll
    for (int q = 0; q < 2; ++q) {
      int s = tid + q * 256;                 // 512 int4 slots
      int r = s >> 4, c = (s & 15) * 4;
      const int* gsrc = adj + (size_t)(ibase + r) * N_NODES + jbase + c;
#if USE_ASYNC_LDS
      ASYNC_CP16(gsrc, &adjT[buf][r][c]);
#else
      *(int4*)(&adjT[buf][r][c]) = *(const int4*)gsrc;
#endif
    }
#pragma unroll
    for (int q = 0; q < 4; ++q) {
      int s = tid + q * 256;                 // 1024 float4 slots
      int r = s >> 4, c = (s & 15) * 4;
      const float* gsrc = H + (size_t)(jbase + r) * F_OUT + c;
#if USE_ASYNC_LDS
      ASYNC_CP16(gsrc, &hT[buf][r][c]);
#else
      *(float4*)(&hT[buf][r][c]) = *(const float4*)gsrc;
#endif
    }
  };

  if (tid < TM) {
    float e = s1[ibase + tid] + *s2max;
    mT[tid] = (e > 0.f) ? e : ALPHA * e;
  }

  // p-phase ownership: thread -> (prow, 8 consecutive cols)
  const int   prow  = tid >> 3;
  const int   pcol0 = (tid & 7) * 8;
  const float s1r   = s1[ibase + prow];

  issue_tile(0, 0);                          // preload tile 0
  __syncthreads();                           // mT visible
  const float mr = mT[prow];

  v8f   acc  = {};
  float zacc = 0.f;

  const int NT = N_NODES / TJ;
  for (int t = 0; t < NT; ++t) {
    const int cur   = t & 1;
    const int jbase = t * TJ;

    if (t + 1 < NT) {
      issue_tile(cur ^ 1, jbase + TJ);       // prefetch next tile
      WAIT_ASYNC(6);                         // older (current) tile complete
    } else {
      WAIT_ASYNC(0);
    }
    __syncthreads();                         // all waves' DMA visible

    // ---- p = adj>0 ? exp(leakyrelu(s1_i + s2_j) - m_i) : 0   (+ Z partials)
#pragma unroll
    for (int q = 0; q < 8; ++q) {
      int   c = pcol0 + q;
      float e = s1r + s2[jbase + c];
      e = (e > 0.f) ? e : ALPHA * e;
      float p = (adjT[cur][prow][c] > 0) ? __expf(e - mr) : 0.f;
      pT[prow][c] = p;
      zacc += p;
    }
    __syncthreads();

    // ---- acc(16x16) += P(16xTJ) @ Htile(TJ x 16-slice)  via f32 WMMA
#pragma unroll
    for (int k0 = 0; k0 < TJ; k0 += 4) {
      v2f a, b;
      a.x = pT[msub * 16 + lm][k0 + kof];
      a.y = pT[msub * 16 + lm][k0 + kof + 1];
      b.x = hT[cur][k0 + kof    ][ntile * 16 + lm];
      b.y = hT[cur][k0 + kof + 1][ntile * 16 + lm];
      acc = __builtin_amdgcn_wmma_f32_16x16x4_f32(false, a, false, b,
                                                  (short)0, acc, false, false);
    }
    __syncthreads();                         // before pT / buffers are reused
  }

  // ---- softmax denominator Z_i (deterministic LDS reduce)
  zscr[prow][tid & 7] = zacc;
  __syncthreads();
  if (tid < TM) {
    float z = 0.f;
#pragma unroll
    for (int q = 0; q < 8; ++q) z += zscr[tid][q];
    Zrow[tid] = z;
  }
  __syncthreads();

  // ---- h' = acc / Z
  const int rloc = msub * 16 + ((lane < 16) ? 0 : 8);
  const int col  = ntile * 16 + lm;
#pragma unroll
  for (int r = 0; r < 8; ++r) {
    int   rr  = rloc + r;
    float z   = Zrow[rr];
    float inv = (z > 0.f) ? 1.f / z : 0.f;   // empty row -> 0 (never w/ p~0.5 adj)
    out[(size_t)(ibase + rr) * F_OUT + col] = acc[r] * inv;
  }
}

// ---------------------------------------------------------------------------
// Launcher. ws layout: H (2 MiB) | s1 (32 KiB) | s2 (32 KiB) | blockmax | s2max
// ---------------------------------------------------------------------------
extern "C" void kernel_launch(void* const* d_in, const int* in_sizes, int n_in,
                              void* d_out, int out_size, void* d_ws, size_t ws_size,
                              hipStream_t stream) {
  (void)in_sizes; (void)n_in; (void)out_size; (void)ws_size;
  const float* X   = (const float*)d_in[0];
  const int*   adj = (const int*)d_in[1];
  const float* W   = (const float*)d_in[2];
  const float* a   = (const float*)d_in[3];
  float*       out = (float*)d_out;

  char*  ws  = (char*)d_ws;
  float* H   = (float*)(ws);                                  // 8192*64*4
  float* s1  = (float*)(ws + 2u * 1024 * 1024);
  float* s2  = (float*)(ws + 2u * 1024 * 1024 + 32 * 1024);
  float* bmx = (float*)(ws + 2u * 1024 * 1024 + 64 * 1024);
  float* smx = (float*)(ws + 2u * 1024 * 1024 + 64 * 1024 + 256);

  hipLaunchKernelGGL(gat_h_kernel,    dim3(N_NODES / 16),  dim3(128), 0, stream, X, W, H);
  hipLaunchKernelGGL(gat_s_kernel,    dim3(N_NODES / 256), dim3(256), 0, stream, H, a, s1, s2, bmx);
  hipLaunchKernelGGL(gat_smax_kernel, dim3(1),             dim3(32),  0, stream, bmx, smx);
  hipLaunchKernelGGL(gat_attn_kernel, dim3(N_NODES / TM),  dim3(256), 0, stream, H, adj, s1, s2, smx, out);
}